// Model_21431886807852
// MI455X (gfx1250) — compile-verified
//
#include <hip/hip_runtime.h>

typedef float v2f __attribute__((ext_vector_type(2)));
typedef float v8f __attribute__((ext_vector_type(8)));

#define NBR 32
#define DIM 64
#define KSEL 5
#define MST 68   // LDS row stride (floats) for msgs: 4*row spread -> conflict-free b64 operand loads
#define SST 33   // LDS row stride (floats) for sims: odd stride -> conflict-free column walks

__global__ __launch_bounds__(32) void submod_select_kernel(
    const float* __restrict__ mail,
    const float* __restrict__ norm_src,
    const float* __restrict__ norm_dst,
    float* __restrict__ out, int B)
{
  __shared__ float sm_msgs[NBR * MST];   // 8704 B
  __shared__ float sm_sims[NBR * SST];   // 4224 B
  __shared__ float sm_cache[NBR];        //  128 B

  const int lane = threadIdx.x;          // 0..31 (wave32)
  const int b = blockIdx.x;
  if (b >= B) return;

  const size_t base = (size_t)b * (NBR * DIM);

  // ---- msgs = mail * norm_src, coalesced float4 global loads -> LDS ----
  float nsl = norm_src[(size_t)b * NBR + lane];   // lane holds norm_src[b][lane]
  #pragma unroll
  for (int it = 0; it < 16; ++it) {
    int e = it * 128 + lane * 4;                  // flat element in 32x64 tile
    float4 v = *(const float4*)(mail + base + e);
    int row = e >> 6;
    int col = e & 63;
    float ns = __shfl(nsl, row, 32);
    v.x *= ns; v.y *= ns; v.z *= ns; v.w *= ns;
    *(float4*)(&sm_msgs[row * MST + col]) = v;    // 16B aligned (MST even, col%4==0)
  }
  __syncthreads();   // single-wave workgroup: S_NOP; LDS is in-order within a wave

  // ---- Gram G = X * X^T via V_WMMA_F32_16X16X4_F32 ----
  // f32 A (16x4) and B (4x16 of X^T) have identical lane/VGPR images:
  // lane L holds X[blk*16 + (L&15)][4s + 2*(L>>4) .. +1]
  const int half = lane >> 4;
  const int r16  = lane & 15;
  v8f g00 = {}; v8f g01 = {}; v8f g10 = {}; v8f g11 = {};
  #pragma unroll
  for (int s = 0; s < 16; ++s) {
    int off = 4 * s + 2 * half;
    v2f p0 = *(const v2f*)(&sm_msgs[r16 * MST + off]);         // rows 0..15 K-slice
    v2f p1 = *(const v2f*)(&sm_msgs[(16 + r16) * MST + off]);  // rows 16..31 K-slice
    g00 = __builtin_amdgcn_wmma_f32_16x16x4_f32(false, p0, false, p0, (short)0, g00, false, false);
    g01 = __builtin_amdgcn_wmma_f32_16x16x4_f32(false, p0, false, p1, (short)0, g01, false, false);
    g10 = __builtin_amdgcn_wmma_f32_16x16x4_f32(false, p1, false, p0, (short)0, g10, false, false);
    g11 = __builtin_amdgcn_wmma_f32_16x16x4_f32(false, p1, false, p1, (short)0, g11, false, false);
  }

  // ---- spill Gram to LDS (C/D layout: VGPR r -> M = r + 8*half, N = lane&15) ----
  #pragma unroll
  for (int r = 0; r < 8; ++r) {
    int m = r + 8 * half;
    sm_sims[m * SST + r16]              = g00[r];
    sm_sims[m * SST + 16 + r16]         = g01[r];
    sm_sims[(16 + m) * SST + r16]       = g10[r];
    sm_sims[(16 + m) * SST + 16 + r16]  = g11[r];
  }
  __syncthreads();

  // ---- distances (lane = column m), accumulate column sums for the mean ----
  float sqm = sm_sims[lane * SST + lane];          // diag = |x_m|^2 (saved before overwrite)
  float colsum = 0.f;
  for (int n = 0; n < NBR; ++n) {
    float sqn = sm_sims[n * SST + n];              // broadcast read (row n diag still intact)
    float g   = sm_sims[n * SST + lane];
    float d2  = sqn + sqm - 2.f * g;
    float dd  = sqrtf(fmaxf(d2, 0.f));
    sm_sims[n * SST + lane] = dd;                  // store after all loads of this row
    colsum += dd;
  }
  #pragma unroll
  for (int off = 16; off >= 1; off >>= 1)
    colsum += __shfl_xor(colsum, off, 32);
  float mean_d = colsum * (1.0f / (NBR * NBR));
  float scale  = -1.0f / mean_d;                   // SIGMA = 1.0

  // ---- sims = exp(-dist / (sigma*mean)) in place ----
  for (int n = 0; n < NBR; ++n) {
    float dd = sm_sims[n * SST + lane];
    sm_sims[n * SST + lane] = __expf(dd * scale);  // v_exp_f32 path
  }
  sm_cache[lane] = 0.f;
  __syncthreads();

  // ---- greedy k=5 selection; accumulate picked msg rows directly ----
  float acc0 = 0.f, acc1 = 0.f;                    // output dims: lane, lane+32
  for (int it = 0; it < KSEL; ++it) {
    // gain for row n = lane: sum_m max(sims[n,m], cache[m]) - cache[m]
    float gain = 0.f;
    #pragma unroll 8
    for (int m = 0; m < NBR; ++m) {
      float s = sm_sims[lane * SST + m];           // stride-33 rows: bank-conflict-free
      float c = sm_cache[m];                       // broadcast
      gain += fmaxf(s, c) - c;
    }
    // wave argmax, first-max wins (matches jnp.argmax tie-break)
    float bv = gain; int bi = lane;
    #pragma unroll
    for (int off = 16; off >= 1; off >>= 1) {
      float ov = __shfl_xor(bv, off, 32);
      int   oi = __shfl_xor(bi, off, 32);
      if (ov > bv || (ov == bv && oi < bi)) { bv = ov; bi = oi; }
    }
    int sel = bi;                                  // uniform across the wave
    // cache[m] = max(sims[sel,m], cache[m])  (lane = m)
    float c = sm_cache[lane];
    float s = sm_sims[sel * SST + lane];
    sm_cache[lane] = fmaxf(s, c);
    // accumulate picked row (with multiplicity, like the reference)
    acc0 += sm_msgs[sel * MST + lane];
    acc1 += sm_msgs[sel * MST + 32 + lane];
  }

  float nd = norm_dst[b];
  out[(size_t)b * DIM + lane]       = acc0 * nd;
  out[(size_t)b * DIM + 32 + lane]  = acc1 * nd;
}

extern "C" void kernel_launch(void* const* d_in, const int* in_sizes, int n_in,
                              void* d_out, int out_size, void* d_ws, size_t ws_size,
                              hipStream_t stream) {
  const float* mail     = (const float*)d_in[0];   // (B, 32, 64) f32
  const float* norm_src = (const float*)d_in[1];   // (B, 32, 1)  f32
  const float* norm_dst = (const float*)d_in[2];   // (B, 1)      f32
  float* out = (float*)d_out;                      // (B, 64)     f32
  const int B = in_sizes[2];                       // n_dst
  submod_select_kernel<<<B, 32, 0, stream>>>(mail, norm_src, norm_dst, out, B);
}